// AnatomicalTextEnhancer_43250320670912
// MI455X (gfx1250) — compile-verified
//
#include <hip/hip_runtime.h>

typedef __attribute__((ext_vector_type(2))) float v2f;
typedef __attribute__((ext_vector_type(8))) float v8f;
typedef __attribute__((ext_vector_type(4))) unsigned int u32x4;
typedef __attribute__((ext_vector_type(4))) int i32x4;
typedef __attribute__((ext_vector_type(8))) int i32x8;

#define NREG 29
#define NB   32     // batch
#define NH   768    // hidden
#define NP   512    // proj
#define NN   8192   // texts per region
#define NT   128    // text rows per workgroup tile
#define KH   16     // H chunk staged in LDS
#define TPITCH 20   // lds_t row pitch (16 data + 4 pad DWORDs; gcd(20,64)=4 -> conflict-free)
#define NTILES (NN / NT)   // 64
#define NCH (NH / KH)      // 48

#if defined(__has_builtin)
#if __has_builtin(__builtin_amdgcn_tensor_load_to_lds)
#define HAVE_TDM 1
#endif
#endif

// ---------------------------------------------------------------------------
// Kernel 1: vn[r][b][p] = l2norm(visual[b][r+1] @ Wv + bv)   (tiny GEMM, VALU)
// ---------------------------------------------------------------------------
__global__ __launch_bounds__(256)
void k_visual(const float* __restrict__ vis, const float* __restrict__ Wv,
              const float* __restrict__ bv, float* __restrict__ vn)
{
    __shared__ float feat[NH];
    __shared__ float red[256];
    const int blk = blockIdx.x;            // r*32 + b
    const int r = blk >> 5, b = blk & 31;
    const int tid = threadIdx.x;
    const float* src = vis + ((size_t)b * (NREG + 1) + (r + 1)) * NH;
    for (int i = tid; i < NH; i += 256) feat[i] = src[i];
    __syncthreads();
    float o0 = bv[tid], o1 = bv[tid + 256];
    for (int h = 0; h < NH; ++h) {
        const float f = feat[h];
        o0 = fmaf(f, Wv[(size_t)h * NP + tid], o0);
        o1 = fmaf(f, Wv[(size_t)h * NP + tid + 256], o1);
    }
    red[tid] = o0 * o0 + o1 * o1;
    __syncthreads();
    for (int s = 128; s > 0; s >>= 1) {
        if (tid < s) red[tid] += red[tid + s];
        __syncthreads();
    }
    const float inv = 1.0f / fmaxf(sqrtf(red[0]), 1e-12f);
    float* dst = vn + (size_t)blk * NP;
    dst[tid] = o0 * inv;
    dst[tid + 256] = o1 * inv;
}

#if HAVE_TDM
// Issue one TDM descriptor: load a 128x16 f32 tile (row stride NH floats) from
// `gaddr` into LDS at `ldsaddr`, padding each 16-DWORD row to a 20-DWORD pitch.
__device__ __forceinline__ void tdm_load_tile(unsigned ldsaddr, const float* gptr)
{
    const unsigned long long ga = (unsigned long long)(uintptr_t)gptr;
    const u32x4 g0 = {
        1u,                                        // count=1 (valid), gather off
        ldsaddr,                                   // lds_addr
        (unsigned)(ga & 0xffffffffull),            // global_addr[31:0]
        (unsigned)((ga >> 32) & 0x01ffffffull) | (2u << 30)  // addr[56:32] | type=2
    };
    const i32x8 g1 = {
        (int)((2u << 16) | (1u << 20) | (3u << 22) | (3u << 25)),
        //  data_size=4B | pad_enable | pad_interval=16DW | pad_amount=4DW; wg_mask=0
        (int)(((unsigned)NH & 0xffffu) << 16),     // tensor_dim0[15:0]=768
        (int)(((unsigned)NN & 0xffffu) << 16),     // tensor_dim0 hi=0 | tensor_dim1[15:0]=8192
        (int)((unsigned)KH << 16),                 // tensor_dim1 hi=0 | tile_dim0=16
        (int)NT,                                   // tile_dim1=128 | tile_dim2=0
        (int)NH,                                   // tensor_dim0_stride[31:0]=768
        0,                                         // stride hi | tensor_dim1_stride lo
        0
    };
    const i32x4 z4 = {};
#if __clang_major__ >= 23
    const i32x8 z8 = {};
    __builtin_amdgcn_tensor_load_to_lds(g0, g1, z4, z4, z8, 0);
#else
    __builtin_amdgcn_tensor_load_to_lds(g0, g1, z4, z4, 0);
#endif
}
#endif

// ---------------------------------------------------------------------------
// Kernel 2: per (region, 128-row text tile):
//   Stage A: MT = text_tile @ Wt + bt  (WMMA f32 16x16x4; text streamed by the
//            Tensor Data Mover with double buffering, Wt chunk via async-LDS),
//            in-wave row l2-norms,
//   Stage B: sims = TN_tile @ vn_r^T   (WMMA), mask, max+argmax partials.
// Wave w owns rows w*16..w*16+15 across all 512 output columns.
// ---------------------------------------------------------------------------
__global__ __launch_bounds__(256)
void k_text(const float* __restrict__ text, const float* __restrict__ Wt,
            const float* __restrict__ bt, const float* __restrict__ vn,
            const int* __restrict__ valid_counts,
            unsigned long long* __restrict__ part)
{
    __shared__ __align__(16) float lds_t[2][NT][TPITCH];   // 20.5 KB
    __shared__ __align__(16) union U {
        float w[KH][NP];          // stage A: Wt chunk (32 KB)
        float tn[NT][65];         // stage B: normalized K-chunk (33.3 KB)
    } u;
    __shared__ unsigned long long best[NB];

    const int r    = blockIdx.y;
    const int tile = blockIdx.x;
    const int n0   = tile * NT;
    const int tid  = threadIdx.x;
    const int wave = tid >> 5;
    const int lane = tid & 31;
    const int l16  = lane & 15;
    const int half = lane >> 4;

    const float* tbase = text + ((size_t)r * NN + n0) * NH;

    v8f zero = {};
    v8f acc[32];
#pragma unroll
    for (int t = 0; t < 32; ++t) acc[t] = zero;
    if (tid < NB) best[tid] = 0ull;

#if !HAVE_TDM
    // fallback staging geometry: row = tid/2 (wave w stages the rows it consumes)
    const int trow = tid >> 1;
    const int tcol = (tid & 1) * 8;
#endif

    // ---- prologue: stage text chunk 0 into buffer 0 ------------------------
#if HAVE_TDM
    if (wave == 0)
        tdm_load_tile((unsigned)(uintptr_t)&lds_t[0][0][0], tbase);
#else
    {
        const float* src = tbase + (size_t)trow * NH + tcol;
        unsigned dst = (unsigned)(uintptr_t)&lds_t[0][trow][tcol];
        asm volatile("global_load_async_to_lds_b128 %0, %1, off"
                     :: "v"(dst), "v"(src) : "memory");
        asm volatile("global_load_async_to_lds_b128 %0, %1, off"
                     :: "v"(dst + 16u), "v"(src + 4) : "memory");
    }
#endif

    // ---------------- Stage A: MT(128x512) = T(128x768) @ Wt ----------------
    for (int it = 0; it < NCH; ++it) {
        const int h0  = it * KH;
        const int cur = it & 1;
        __syncthreads();   // prior iter's u.w / lds_t[cur^1] reads complete

        // async-stage Wt chunk: 16x512 floats, 8 x B128 per thread
#pragma unroll
        for (int j = 0; j < 8; ++j) {
            const int f0  = (tid + j * 256) * 4;
            const int row = f0 >> 9;
            const int col = f0 & 511;
            const float* wsrc = &Wt[(size_t)(h0 + row) * NP + col];
            unsigned wdst = (unsigned)(uintptr_t)&u.w[row][col];
            asm volatile("global_load_async_to_lds_b128 %0, %1, off"
                         :: "v"(wdst), "v"(wsrc) : "memory");
        }

#if HAVE_TDM
        // prefetch next text chunk via TDM into the other buffer
        if (wave == 0 && it + 1 < NCH)
            tdm_load_tile((unsigned)(uintptr_t)&lds_t[cur ^ 1][0][0],
                          tbase + (h0 + KH));
        asm volatile("s_wait_asynccnt 0x0" ::: "memory");   // Wt chunk landed
        if (wave == 0) {
            if (it + 1 < NCH) __builtin_amdgcn_s_wait_tensorcnt(1);
            else              __builtin_amdgcn_s_wait_tensorcnt(0);
        }
#else
        if (it + 1 < NCH) {
            const float* src = tbase + (size_t)trow * NH + (h0 + KH) + tcol;
            unsigned dst = (unsigned)(uintptr_t)&lds_t[cur ^ 1][trow][tcol];
            asm volatile("global_load_async_to_lds_b128 %0, %1, off"
                         :: "v"(dst), "v"(src) : "memory");
            asm volatile("global_load_async_to_lds_b128 %0, %1, off"
                         :: "v"(dst + 16u), "v"(src + 4) : "memory");
            asm volatile("s_wait_asynccnt 0x2" ::: "memory");
        } else {
            asm volatile("s_wait_asynccnt 0x0" ::: "memory");
        }
#endif
        __syncthreads();   // W chunk + current text tile visible to all waves

#pragma unroll
        for (int kk = 0; kk < KH; kk += 4) {
            v2f a;
            a.x = lds_t[cur][wave * 16 + l16][kk + half * 2];
            a.y = lds_t[cur][wave * 16 + l16][kk + half * 2 + 1];
#pragma unroll
            for (int t = 0; t < 32; ++t) {
                const int col = t * 16 + l16;
                v2f bf;
                bf.x = u.w[kk + half * 2][col];
                bf.y = u.w[kk + half * 2 + 1][col];
                acc[t] = __builtin_amdgcn_wmma_f32_16x16x4_f32(
                    false, a, false, bf, (short)0, acc[t], false, false);
            }
        }
    }

    // ---- bias + in-wave row l2-norms (rows are wave-private) ---------------
    float invnv[8];
    {
        float ss[8];
#pragma unroll
        for (int v = 0; v < 8; ++v) ss[v] = 0.0f;
#pragma unroll
        for (int t = 0; t < 32; ++t) {
            const float bias = bt[t * 16 + l16];
#pragma unroll
            for (int v = 0; v < 8; ++v) {
                acc[t][v] += bias;
                ss[v] = fmaf(acc[t][v], acc[t][v], ss[v]);
            }
        }
#pragma unroll
        for (int v = 0; v < 8; ++v) {
            float sv = ss[v];
            sv += __shfl_xor(sv, 1);
            sv += __shfl_xor(sv, 2);
            sv += __shfl_xor(sv, 4);
            sv += __shfl_xor(sv, 8);   // butterfly: every lane holds row sum
            invnv[v] = 1.0f / fmaxf(sqrtf(sv), 1e-12f);
        }
    }

    // ---------------- Stage B: sims(128x32) = TN @ vn_r^T -------------------
    const float* vnr = vn + (size_t)r * NB * NP;
    v8f sim0 = zero, sim1 = zero;

    for (int kc = 0; kc < NP; kc += 64) {
        __syncthreads();   // previous chunk reads (or last u.w use) done
        const int t0 = kc >> 4;
#pragma unroll
        for (int tt = 0; tt < 4; ++tt) {
#pragma unroll
            for (int v = 0; v < 8; ++v) {
                const int row = wave * 16 + v + half * 8;
                u.tn[row][tt * 16 + l16] = acc[t0 + tt][v] * invnv[v];
            }
        }
        __syncthreads();
#pragma unroll 4
        for (int k = 0; k < 64; k += 4) {
            v2f a;
            a.x = u.tn[wave * 16 + l16][k + half * 2];
            a.y = u.tn[wave * 16 + l16][k + half * 2 + 1];
            const float* v0 = vnr + (size_t)l16 * NP + kc + k + half * 2;
            v2f b0; b0.x = v0[0]; b0.y = v0[1];
            sim0 = __builtin_amdgcn_wmma_f32_16x16x4_f32(
                false, a, false, b0, (short)0, sim0, false, false);
            const float* v1 = v0 + 16 * NP;
            v2f b1; b1.x = v1[0]; b1.y = v1[1];
            sim1 = __builtin_amdgcn_wmma_f32_16x16x4_f32(
                false, a, false, b1, (short)0, sim1, false, false);
        }
    }

    // ---------------- mask + per-b max/argmax over this 128-row tile --------
    __syncthreads();
    const int vc = valid_counts[r];
    unsigned long long p0 = 0ull, p1 = 0ull;
#pragma unroll
    for (int v = 0; v < 8; ++v) {
        const int ng = n0 + wave * 16 + v + half * 8;
        const unsigned nkey = (unsigned)(~(unsigned)ng);
        {
            const float sv = (ng < vc) ? sim0[v] : -1.0e9f;
            const unsigned ub = __float_as_uint(sv);
            const unsigned key = (ub & 0x80000000u) ? ~ub : (ub | 0x80000000u);
            const unsigned long long pk = ((unsigned long long)key << 32) | nkey;
            p0 = (pk > p0) ? pk : p0;
        }
        {
            const float sv = (ng < vc) ? sim1[v] : -1.0e9f;
            const unsigned ub = __float_as_uint(sv);
            const unsigned key = (ub & 0x80000000u) ? ~ub : (ub | 0x80000000u);
            const unsigned long long pk = ((unsigned long long)key << 32) | nkey;
            p1 = (pk > p1) ? pk : p1;
        }
    }
    atomicMax(&best[l16], p0);
    atomicMax(&best[16 + l16], p1);
    __syncthreads();
    if (tid < NB)
        part[((size_t)r * NTILES + tile) * NB + tid] = best[tid];
}

// ---------------------------------------------------------------------------
// Kernel 3: fold 64 tile partials -> best_sim (f32) + best_idx (i32)
// ---------------------------------------------------------------------------
__global__ void k_reduce(const unsigned long long* __restrict__ part,
                         float* __restrict__ out_sim, int* __restrict__ out_idx)
{
    const int r = blockIdx.x;
    const int b = threadIdx.x;   // 32 threads
    unsigned long long m = 0ull;
    const unsigned long long* p = part + (size_t)r * NTILES * NB + b;
    for (int t = 0; t < NTILES; ++t) {
        const unsigned long long v = p[(size_t)t * NB];
        m = (v > m) ? v : m;
    }
    const unsigned key = (unsigned)(m >> 32);
    const unsigned ub = (key & 0x80000000u) ? (key ^ 0x80000000u) : ~key;
    const int idx = (int)(~((unsigned)(m & 0xffffffffu)));
    out_sim[b * NREG + r] = __uint_as_float(ub);
    out_idx[b * NREG + r] = idx;
}

// ---------------------------------------------------------------------------
extern "C" void kernel_launch(void* const* d_in, const int* in_sizes, int n_in,
                              void* d_out, int out_size, void* d_ws, size_t ws_size,
                              hipStream_t stream) {
    const float* vis  = (const float*)d_in[0];
    const float* text = (const float*)d_in[1];
    const float* Wv   = (const float*)d_in[2];
    const float* bv   = (const float*)d_in[3];
    const float* Wt   = (const float*)d_in[4];
    const float* bt   = (const float*)d_in[5];
    const int*   vc   = (const int*)d_in[6];

    char* ws = (char*)d_ws;
    float* vn = (float*)ws;                                   // 29*32*512 f32
    const size_t vn_bytes = (size_t)NREG * NB * NP * sizeof(float);
    unsigned long long* part = (unsigned long long*)(ws + vn_bytes);

    float* out_sim = (float*)d_out;                           // (32,29)
    int*   out_idx = (int*)((float*)d_out + NB * NREG);       // (32,29)

    k_visual<<<dim3(NREG * NB), 256, 0, stream>>>(vis, Wv, bv, vn);
    k_text<<<dim3(NTILES, NREG), 256, 0, stream>>>(text, Wt, bt, vn, vc, part);
    k_reduce<<<dim3(NREG), 32, 0, stream>>>(part, out_sim, out_idx);
}